// NeighborsConvolution_1451698946407
// MI455X (gfx1250) — compile-verified
//
#include <hip/hip_runtime.h>

typedef float v2f __attribute__((ext_vector_type(2)));
typedef float v8f __attribute__((ext_vector_type(8)));
typedef unsigned int u32x4 __attribute__((ext_vector_type(4)));
typedef int i32x4 __attribute__((ext_vector_type(4)));
typedef int i32x8 __attribute__((ext_vector_type(8)));

#define NB    8
#define NPTS  2048
#define CIN   64
#define COUT  64
#define KB    64                  // b-block staged per iteration
#define ATILE 16                  // a rows per workgroup
#define TILE_ELEMS (KB * CIN)     // 4096 f32 = 16 KB per staged tile

#if defined(__has_builtin)
#  if __has_builtin(__builtin_amdgcn_tensor_load_to_lds) && \
      __has_builtin(__builtin_amdgcn_s_wait_tensorcnt)
#    define USE_TDM 1
#  endif
#endif
#ifndef USE_TDM
#  define USE_TDM 0
#endif

#if USE_TDM
// 1-D contiguous tile copy via the Tensor Data Mover (D# per CDNA5 ISA §8):
//   group0: count=1 | lds_addr | global_addr(57b) | type=2
//   group1: data_size=4B, tensor_dim0=tile_dim0=4096, tensor_dim1=tile_dim1=1
// This toolchain exposes the 6-arg builtin:
//   (u32x4 g0, i32x8 g1, i32x4 g2, i32x4 g3, i32x8 extra, i32 cpol)
__device__ __forceinline__ void tdm_load_f_tile(const float* gsrc,
                                                unsigned lds_byte_off) {
  unsigned long long ga = (unsigned long long)(uintptr_t)gsrc;
  u32x4 g0;
  g0[0] = 1u;                                               // count=1, user D#
  g0[1] = lds_byte_off;                                     // lds_addr [63:32]
  g0[2] = (unsigned)(ga & 0xffffffffu);                     // global_addr lo
  g0[3] = (unsigned)((ga >> 32) & 0x1ffffffu) | (2u << 30); // ga[56:32] | type=2
  i32x8 g1;
  g1[0] = (int)(2u << 16);                         // wg_mask=0, data_size=2 (4B)
  g1[1] = (int)((TILE_ELEMS & 0xffff) << 16);      // tensor_dim0[15:0] @ [63:48]
  g1[2] = (int)(((TILE_ELEMS >> 16) & 0xffff) |    // tensor_dim0[31:16]
                (1 << 16));                        // tensor_dim1[15:0] = 1
  g1[3] = (int)((TILE_ELEMS & 0xffff) << 16);      // tensor_dim1 hi=0 | tile_dim0
  g1[4] = 1;                                       // tile_dim1=1, tile_dim2=0
  g1[5] = (int)TILE_ELEMS;                         // tensor_dim0_stride lo
  g1[6] = 0;                                       // stride hi | dim1_stride lo
  g1[7] = 0;
  i32x4 zz4 = {0, 0, 0, 0};                        // groups 2/3 unused (<=2D)
  i32x8 zz8 = {0, 0, 0, 0, 0, 0, 0, 0};
  __builtin_amdgcn_tensor_load_to_lds(g0, g1, zz4, zz4, zz8, 0);
}
#endif

// out[z,a,i] = sum_x sum_j ( sum_b mask(a,b)*diff_x(a,b)*feat[b,j] ) * Wk[x,i,j]
// Stage 1: per x, S_x = A_x @ F with A_x[a,b] = mask*(g[b,x]-g[a,x]) built in LDS.
// Stage 2: out = S (16 x 192) @ W2 (192 x 64), W2[(x,j), i] = Wk[x,i,j].
__global__ __launch_bounds__(128)
void neighbors_conv_wmma(const float* __restrict__ feat,
                         const float* __restrict__ geom,
                         const float* __restrict__ Wk,
                         float* __restrict__ out) {
  __shared__ __attribute__((aligned(16))) float lds_f[2][TILE_ELEMS]; // 32 KB dbuf
  __shared__ float lds_md[3 * ATILE * KB];   // md[x][a][b]; reused as s[a][192]
  __shared__ float lds_ga[ATILE * 3];        // g of the 16 'a' rows
  __shared__ float lds_gb[KB * 3];           // g of the 64 'b' rows

  const int tid  = threadIdx.x;
  const int wave = tid >> 5;                 // 0..3 -> j/i tile
  const int lane = tid & 31;
  const int l    = lane & 15;
  const int hi   = lane >> 4;                // lane half selects K pair / M half

  const int z  = blockIdx.y;
  const int a0 = blockIdx.x * ATILE;

  const float* f = feat + (size_t)z * NPTS * CIN;
  const float* g = geom + (size_t)z * NPTS * 3;

  // geometry of the 16 output rows (constant across the b loop)
  for (int i = tid; i < ATILE * 3; i += 128)
    lds_ga[i] = g[a0 * 3 + i];

#if USE_TDM
  // Prefetch block 0 into buffer 0 (one DMA from wave 0; EXEC==0 waves skip).
  if (wave == 0)
    tdm_load_f_tile(f, (unsigned)(uintptr_t)(&lds_f[0][0]));
#endif

  v8f acc0 = {}, acc1 = {}, acc2 = {};

  const int NBLK = NPTS / KB;  // 32
  for (int ib = 0; ib < NBLK; ++ib) {
    const int buf = ib & 1;
    __syncthreads();  // GEMM(ib-1) done reading LDS -> buffer buf^1 is free

#if USE_TDM
    if (wave == 0) {
      if (ib + 1 < NBLK) {
        // overlap: DMA next tile while this block is consumed
        tdm_load_f_tile(f + (size_t)(ib + 1) * TILE_ELEMS,
                        (unsigned)(uintptr_t)(&lds_f[buf ^ 1][0]));
        __builtin_amdgcn_s_wait_tensorcnt(1);  // in-order: tile ib has landed
      } else {
        __builtin_amdgcn_s_wait_tensorcnt(0);  // last tile
      }
    }
#else
    {
      const float4* fsrc = (const float4*)(f + (size_t)ib * TILE_ELEMS);
      float4* fdst = (float4*)&lds_f[buf][0];
      #pragma unroll
      for (int i = tid; i < TILE_ELEMS / 4; i += 128)
        fdst[i] = fsrc[i];
    }
#endif
    for (int i = tid; i < KB * 3; i += 128)
      lds_gb[i] = g[ib * (KB * 3) + i];
    __syncthreads();  // tile ib + g block visible to all waves

    // ---- build mask*diff for all (a,b,x), shared by every wave/x-GEMM ----
    #pragma unroll
    for (int q = 0; q < (ATILE * KB) / 128; ++q) {
      int pi  = q * 128 + tid;       // pi = a_l*64 + b_l
      int a_l = pi >> 6;
      int b_l = pi & 63;
      float dx = lds_gb[b_l * 3 + 0] - lds_ga[a_l * 3 + 0];
      float dy = lds_gb[b_l * 3 + 1] - lds_ga[a_l * 3 + 1];
      float dz = lds_gb[b_l * 3 + 2] - lds_ga[a_l * 3 + 2];
      float n2 = dx * dx + dy * dy + dz * dz;
      float m  = (n2 < 1.0f) ? 1.0f : 0.0f;   // norm<1  <=>  norm^2<1
      lds_md[0 * (ATILE * KB) + pi] = m * dx;
      lds_md[1 * (ATILE * KB) + pi] = m * dy;
      lds_md[2 * (ATILE * KB) + pi] = m * dz;
    }
    __syncthreads();

    // ---- WMMA GEMM: this wave owns j-tile [j0, j0+16) ----
    const int j0 = wave * 16;
    const float* ftile = &lds_f[buf][0];
    #pragma unroll
    for (int t = 0; t < KB / 4; ++t) {
      const int bb = 4 * t + 2 * hi;  // K pair for this lane half (even -> b64)
      // A frags (16x4 f32): lane<16 holds K=bb,bb+1 of row M=l; lanes>=16 K+2
      v2f aX = *(const v2f*)&lds_md[0 * (ATILE * KB) + l * KB + bb];
      v2f aY = *(const v2f*)&lds_md[1 * (ATILE * KB) + l * KB + bb];
      v2f aZ = *(const v2f*)&lds_md[2 * (ATILE * KB) + l * KB + bb];
      // B frag (4x16 f32): col N = j0+l, rows K = bb, bb+1
      v2f bF;
      bF.x = ftile[bb * CIN + j0 + l];
      bF.y = ftile[(bb + 1) * CIN + j0 + l];
      acc0 = __builtin_amdgcn_wmma_f32_16x16x4_f32(false, aX, false, bF, (short)0, acc0, false, false);
      acc1 = __builtin_amdgcn_wmma_f32_16x16x4_f32(false, aY, false, bF, (short)0, acc1, false, false);
      acc2 = __builtin_amdgcn_wmma_f32_16x16x4_f32(false, aZ, false, bF, (short)0, acc2, false, false);
    }
  }

  // ---- park S[a, p] (p = x*64 + j) in LDS, reusing lds_md (16*192 floats) ----
  __syncthreads();
  float* lds_s = lds_md;
  {
    const int j0 = wave * 16;
    #pragma unroll
    for (int r = 0; r < 8; ++r) {
      const int a = r + 8 * hi;   // C/D layout: vgpr r, lane half hi -> row r+8*hi
      lds_s[a * 192 + 0 * 64 + j0 + l] = acc0[r];
      lds_s[a * 192 + 1 * 64 + j0 + l] = acc1[r];
      lds_s[a * 192 + 2 * 64 + j0 + l] = acc2[r];
    }
  }
  __syncthreads();

  // ---- stage 2: out[a, i] = sum_p S[a,p] * Wk[p/64, i, p%64], K = 192 ----
  v8f accO = {};
  const int i0 = wave * 16;
  #pragma unroll
  for (int t = 0; t < 48; ++t) {
    const int pb = 4 * t + 2 * hi;
    v2f aS = *(const v2f*)&lds_s[l * 192 + pb];      // 8B aligned: pb even
    const int x  = pb >> 6;
    const int jj = pb & 63;                          // jj even; jj,jj+1 same x
    const float2 w2 = *(const float2*)&Wk[x * (COUT * CIN) + (i0 + l) * CIN + jj];
    v2f bW; bW.x = w2.x; bW.y = w2.y;
    accO = __builtin_amdgcn_wmma_f32_16x16x4_f32(false, aS, false, bW, (short)0, accO, false, false);
  }

  // N_NORM == 1 -> no scaling. Coalesced stores: lanes 0..15 cover i0..i0+15.
  float* op = out + ((size_t)z * NPTS + a0) * COUT;
  #pragma unroll
  for (int r = 0; r < 8; ++r)
    op[(r + 8 * hi) * COUT + i0 + l] = accO[r];
}

extern "C" void kernel_launch(void* const* d_in, const int* in_sizes, int n_in,
                              void* d_out, int out_size, void* d_ws, size_t ws_size,
                              hipStream_t stream) {
  (void)in_sizes; (void)n_in; (void)d_ws; (void)ws_size; (void)out_size;
  const float* feat = (const float*)d_in[0];
  const float* geom = (const float*)d_in[1];
  const float* Wk   = (const float*)d_in[2];
  float* out = (float*)d_out;

  dim3 grid(NPTS / ATILE, NB);   // 128 a-tiles x 8 batches
  dim3 block(128);               // 4 waves: one 16-wide j/i tile each
  neighbors_conv_wmma<<<grid, block, 0, stream>>>(feat, geom, Wk, out);
}